// iFreqProj_80049600463646
// MI455X (gfx1250) — compile-verified
//
#include <hip/hip_runtime.h>
#include <math.h>

// ---------------------------------------------------------------------------
// iFreqProj on MI455X (gfx1250, wave32).
//
// u[b,n] = (1/F^2) Re[ sum_i p_b[i] * sum_j fc[n,i,j] * q_b[j] ]
//   fc = mean_b f  (complex),  p_b[i]=exp(i*2pi*x_b0*i),  q_b[j]=exp(i*2pi*x_b1*j)
//
// Pass 1 streams the 134 MB of f once (memory-bound, NT loads).
// Pass 3 does the j-contraction as an fp32 WMMA GEMM:
//   A (1024x1024) = mean(f) reinterpreted row-major: row m=n*512+i, col k=2j+c
//   B (1024x64):   col 2b   = [qr_b[j]; -qi_b[j]]  -> S[m,2b]   = Re part
//                  col 2b+1 = [qi_b[j];  qr_b[j]]  -> S[m,2b+1] = Im part
// ---------------------------------------------------------------------------

typedef __attribute__((ext_vector_type(2))) float v2f;
typedef __attribute__((ext_vector_type(4))) float v4f;
typedef __attribute__((ext_vector_type(8))) float v8f;

#define NRFFT   512
#define BATCH   32
#define MDIM    1024              // 2 * 512 rows (n,i)
#define KDIM    1024              // 2 * 512 cols (j, re/im interleaved)
#define NDIM    64                // 2 * 32 cols (b, re/im)
#define FM_ELEMS (MDIM * KDIM)    // 1,048,576 floats (4 MB)

// ---- Kernel 1: fm = mean over batch of f. Streams 134 MB (the hot pass). ---
__global__ void kmean_batch(const float* __restrict__ f, float* __restrict__ fm) {
    int idx = blockIdx.x * blockDim.x + threadIdx.x;        // 262,144 float4s
    const v4f* f4 = (const v4f*)f;
    v4f acc = {0.f, 0.f, 0.f, 0.f};
#pragma unroll
    for (int b = 0; b < BATCH; ++b) {
        v4f v = __builtin_nontemporal_load(&f4[(size_t)b * (FM_ELEMS / 4) + idx]);
        acc += v;
    }
    acc *= (1.0f / (float)BATCH);
    ((v4f*)fm)[idx] = acc;
}

// ---- Kernel 2: build phase B-matrix (KDIM x NDIM) and p vectors. -----------
__global__ void kbuild_phase(const float* __restrict__ x,
                             float* __restrict__ Bm,      // KDIM*NDIM floats
                             float* __restrict__ P) {     // [pr(16384) | pi(16384)]
    int t = blockIdx.x * blockDim.x + threadIdx.x;        // 32*512 threads
    int b = t >> 9;
    int j = t & 511;
    float x0 = x[b * 2 + 0];
    float x1 = x[b * 2 + 1];
    const float TWO_PI = 6.28318530717958647692f;

    float qs, qc;
    sincosf(TWO_PI * x1 * (float)j, &qs, &qc);
    Bm[(2 * j)     * NDIM + 2 * b]     = qc;   // sr column:  fr*qr - fi*qi
    Bm[(2 * j + 1) * NDIM + 2 * b]     = -qs;
    Bm[(2 * j)     * NDIM + 2 * b + 1] = qs;   // si column:  fr*qi + fi*qr
    Bm[(2 * j + 1) * NDIM + 2 * b + 1] = qc;

    float ps, pc;
    sincosf(TWO_PI * x0 * (float)j, &ps, &pc); // j plays the role of i here
    P[b * NRFFT + j]                 = pc;     // pr
    P[BATCH * NRFFT + b * NRFFT + j] = ps;     // pi
}

// ---- Kernel 3: S = A * B via V_WMMA_F32_16X16X4_F32 tiles. -----------------
// One wave per 16x16 output tile. 64 row-tiles x 4 col-tiles = 256 waves.
// A 16x4 layout : lane L holds A[M=L%16, K=2*(L/16)+{0,1}] in a.{x,y}
// B 4x16 layout : lane L holds B[K=2*(L/16)+{0,1}, N=L%16] in b.{x,y}
// C 16x16 layout: VGPR r, lane L holds C[M=r+8*(L/16), N=L%16]
__global__ void kgemm_wmma(const float* __restrict__ A,
                           const float* __restrict__ Bm,
                           float* __restrict__ S) {
    int wave = (blockIdx.x * blockDim.x + threadIdx.x) >> 5;
    int lane = threadIdx.x & 31;
    int mt = wave >> 2;                 // 0..63
    int nt = wave & 3;                  // 0..3
    int l16   = lane & 15;
    int khalf = (lane >> 4) << 1;       // 0 or 2

    int mrow = mt * 16 + l16;           // A row for this lane
    int ncol = nt * 16 + l16;           // B/S column for this lane

    const float* arow = A + (size_t)mrow * KDIM;

    v8f c = {0.f, 0.f, 0.f, 0.f, 0.f, 0.f, 0.f, 0.f};
    for (int k0 = 0; k0 < KDIM; k0 += 4) {
        v2f a = *(const v2f*)(arow + k0 + khalf);          // 8B aligned
        const float* bp = Bm + (size_t)(k0 + khalf) * NDIM + ncol;
        v2f b;
        b.x = bp[0];
        b.y = bp[NDIM];
        c = __builtin_amdgcn_wmma_f32_16x16x4_f32(
                /*neg_a=*/false, a, /*neg_b=*/false, b,
                /*c_mod=*/(short)0, c, /*reuse_a=*/false, /*reuse_b=*/false);
    }

    int mbase = mt * 16 + 8 * (lane >> 4);
#pragma unroll
    for (int r = 0; r < 8; ++r)
        S[(size_t)(mbase + r) * NDIM + ncol] = c[r];
}

// ---- Kernel 4: u[b,n] = (1/F^2) * sum_i pr*sr - pi*si. ---------------------
__global__ void kreduce_u(const float* __restrict__ S,
                          const float* __restrict__ P,
                          float* __restrict__ out) {
    int b = blockIdx.x >> 1;
    int n = blockIdx.x & 1;
    int tid = threadIdx.x;              // 256 threads

    float acc = 0.f;
    for (int i = tid; i < NRFFT; i += 256) {
        float sr = S[(size_t)(n * NRFFT + i) * NDIM + 2 * b];
        float si = S[(size_t)(n * NRFFT + i) * NDIM + 2 * b + 1];
        acc += P[b * NRFFT + i] * sr - P[BATCH * NRFFT + b * NRFFT + i] * si;
    }

    __shared__ float red[256];
    red[tid] = acc;
    __syncthreads();
    for (int s = 128; s > 0; s >>= 1) {
        if (tid < s) red[tid] += red[tid + s];
        __syncthreads();
    }
    if (tid == 0)
        out[b * 2 + n] = red[0] * (1.0f / ((float)NRFFT * (float)NRFFT));
}

// ---------------------------------------------------------------------------
extern "C" void kernel_launch(void* const* d_in, const int* in_sizes, int n_in,
                              void* d_out, int out_size, void* d_ws, size_t ws_size,
                              hipStream_t stream) {
    const float* f = (const float*)d_in[0];   // (32, 2, 512, 512, 2) fp32
    const float* x = (const float*)d_in[1];   // (32, 2) fp32
    float* out = (float*)d_out;               // 64 floats

    // Workspace layout (floats):
    float* fm = (float*)d_ws;                       // 1,048,576  (A matrix, 4 MB)
    float* Bm = fm + FM_ELEMS;                      //    65,536  (phase B matrix)
    float* P  = Bm + (size_t)KDIM * NDIM;           //    32,768  (pr | pi)
    float* S  = P + 2 * BATCH * NRFFT;              //    65,536  (GEMM result)

    // 1) batch mean: 262,144 float4 outputs, 1024 blocks x 256 threads
    kmean_batch<<<dim3(FM_ELEMS / 4 / 256), dim3(256), 0, stream>>>(f, fm);

    // 2) phase matrices: 16,384 threads
    kbuild_phase<<<dim3((BATCH * NRFFT) / 256), dim3(256), 0, stream>>>(x, Bm, P);

    // 3) WMMA GEMM: 256 waves = 32 blocks x 256 threads (8 waves/block)
    kgemm_wmma<<<dim3(32), dim3(256), 0, stream>>>(fm, Bm, S);

    // 4) final reduction: one block per (b, n)
    kreduce_u<<<dim3(BATCH * 2), dim3(256), 0, stream>>>(S, P, out);

    (void)in_sizes; (void)n_in; (void)out_size; (void)ws_size;
}